// SimpleSNN_26319559590633
// MI455X (gfx1250) — compile-verified
//
#include <hip/hip_runtime.h>

// ---- problem constants (match reference) ----
#define TSTEPS 256
#define BB     32
#define SS     8192
#define INN    4096
#define HH     512
#define OUTN   10
#define CAP    96        // max retained events per (batch,timestep) bin; E[count]=32
#define ZPAD   516       // LDS row stride (floats): 512 + 4 pad -> bank-conflict-free A reads

typedef float v2f __attribute__((ext_vector_type(2)));
typedef float v8f __attribute__((ext_vector_type(8)));

#if defined(__gfx1250__) && __has_builtin(__builtin_amdgcn_wmma_f32_16x16x4_f32)
#define HAVE_WMMA_F32 1
#endif

// ---------------- setup kernels ----------------

__global__ void snn_zero_kernel(float* p, int n) {
  int g = blockIdx.x * blockDim.x + threadIdx.x;
  if (g < n) p[g] = 0.f;
}

// dst[c*rows + r] = src[r*cols + c]   (coalesced reads)
__global__ void snn_transpose_kernel(const float* __restrict__ src,
                                     float* __restrict__ dst, int rows, int cols) {
  int g = blockIdx.x * blockDim.x + threadIdx.x;
  if (g >= rows * cols) return;
  int r = g / cols, c = g % cols;
  dst[(size_t)c * rows + r] = src[g];
}

// One thread per (batch, timestep): scan that batch's events in s-order
// (preserves last-write-wins of the reference scatter), dedupe by position
// keeping the LAST event, keep only surviving val==1 positions.
__global__ void snn_bin_kernel(const int* __restrict__ x,   // [B,S,4] = cx,cy,val,ts
                               int* __restrict__ binPos,    // [B*T, CAP]
                               int* __restrict__ binCnt) {  // [B*T]
  int g = blockIdx.x * blockDim.x + threadIdx.x;
  if (g >= BB * TSTEPS) return;
  int b   = g / TSTEPS;
  int myt = g % TSTEPS;
  const int* ev = x + (size_t)b * SS * 4;
  int tp[CAP], tv[CAP];
  int cnt = 0;
  for (int s = 0; s < SS; ++s) {              // lanes share b -> broadcast loads
    if (ev[s * 4 + 3] == myt && cnt < CAP) {
      tp[cnt] = ev[s * 4 + 0] * ev[s * 4 + 1];
      tv[cnt] = ev[s * 4 + 2];
      ++cnt;
    }
  }
  int n = 0;
  int* dst = binPos + (size_t)g * CAP;
  for (int k = 0; k < cnt; ++k) {
    bool last = true;
    for (int k2 = k + 1; k2 < cnt; ++k2)
      if (tp[k2] == tp[k]) { last = false; break; }
    if (last && tv[k] == 1) dst[n++] = tp[k];
  }
  binCnt[g] = n;
}

// ---------------- per-timestep kernels ----------------
// K1: 8 blocks x 256 threads.  Block = one M(batch)-tile x 8 N(hidden)-tiles,
// one wave per 16x16 output tile.  The block's 16 spike rows (32 KB) are
// async-copied to LDS once and shared by all 8 waves; B fragments stream from
// the transposed w_recT with fully coalesced 64B half-wave segments.
__global__ void __launch_bounds__(256)
snn_step_k1(const float* __restrict__ w_recT,  // [H,H]  w_recT[k*H + h] = w_rec[h,k]
            const float* __restrict__ w_inT,   // [IN,H]
            const int*   __restrict__ binPos,  // [B*T,CAP]
            const int*   __restrict__ binCnt,  // [B*T]
            const float* __restrict__ z_prev,  // [B,H]
            float*       __restrict__ z_new,   // [B,H]
            float*       __restrict__ v,       // [B,H]
            float*       __restrict__ i_st,    // [B,H]
            int t) {
  __shared__ float zsh[16 * ZPAD];             // 16 spike rows, padded

  const int tid  = threadIdx.x;
  const int lane = tid & 31;
  const int wave = tid >> 5;
  const int half = lane >> 4;                  // 0: lanes 0-15, 1: lanes 16-31
  const int l15  = lane & 15;
  const int b0   = (blockIdx.x & 1) * 16;                 // M tile (batch rows)
  const int h0   = ((blockIdx.x >> 1) * 8 + wave) * 16;   // N tile (hidden cols)

  // ---- stage z_prev[b0 .. b0+15][0..511] (contiguous 32 KB) into LDS ----
  // 8192 16-byte chunks, 32 per thread, consecutive tids -> consecutive chunks.
#if defined(__AMDGCN__)
#pragma unroll
  for (int it = 0; it < 32; ++it) {
    const int c = tid + 256 * it;
    const int r = c >> 7;                      // row: 128 chunks per 512-float row
    const int w = c & 127;
    unsigned ldsOff = (unsigned)(size_t)&zsh[r * ZPAD + w * 4];
    const char* g = (const char*)(z_prev + b0 * HH) + (size_t)c * 16;
    asm volatile("global_load_async_to_lds_b128 %0, %1, off"
                 :: "v"(ldsOff), "v"(g) : "memory");
  }
  asm volatile("s_wait_asynccnt 0" ::: "memory");
#else
  for (int c = tid; c < 16 * 128; c += 256) {
    const int r = c >> 7, w = c & 127;
    for (int e = 0; e < 4; ++e)
      zsh[r * ZPAD + w * 4 + e] = z_prev[(b0 + r) * HH + w * 4 + e];
  }
#endif
  __syncthreads();

#ifdef HAVE_WMMA_F32
  v8f acc = {};
  // D = A x B:  A[m,k] = z_prev[b0+m, k] (from LDS), B[k,n] = w_recT[k, h0+n].
  // f32 16x16x4 fragments (ISA 7.12.2): lanes 0-15 carry K {k0,k0+1},
  // lanes 16-31 carry {k0+2,k0+3}.
  for (int k0 = 0; k0 < HH; k0 += 4) {
    const int ka = k0 + 2 * half;
    v2f a, bf;
    a.x  = zsh[l15 * ZPAD + ka];               // ds_load_b64, conflict-free
    a.y  = zsh[l15 * ZPAD + ka + 1];
    bf.x = w_recT[(size_t)(ka)     * HH + h0 + l15];   // coalesced 64B / half-wave
    bf.y = w_recT[(size_t)(ka + 1) * HH + h0 + l15];
    acc = __builtin_amdgcn_wmma_f32_16x16x4_f32(
        /*neg_a=*/false, a, /*neg_b=*/false, bf,
        /*c_mod=*/(short)0, acc, /*reuse_a=*/false, /*reuse_b=*/false);
  }
#endif

  // C/D layout: VGPR r -> M = r + 8*half, N = lane%16
#pragma unroll
  for (int r = 0; r < 8; ++r) {
    const int bb  = b0 + r + 8 * half;
    const int hh  = h0 + l15;
    const int idx = bb * HH + hh;

    const float vold  = v[idx];
    const float iold  = i_st[idx];
    const float v_dec = vold + 0.1f * (iold - vold);  // DT*TAU_MEM_INV = 0.1
    const float i_dec = 0.8f * iold;                  // 1 - DT*TAU_SYN_INV
    const float z     = (v_dec - 0.5f) > 0.f ? 1.f : 0.f;
    const float v_nw  = (1.f - z) * v_dec;

#ifdef HAVE_WMMA_F32
    const float rec = acc[r];
#else
    float rec = 0.f;
    for (int k = 0; k < HH; ++k)
      rec += zsh[(r + 8 * half) * ZPAD + k] * w_recT[(size_t)k * HH + hh];
#endif

    // sparse event-driven input: sum surviving columns of w_in (transposed)
    float in_c = 0.f;
    const int  bg  = bb * TSTEPS + t;
    const int  cnt = binCnt[bg];
    const int* pl  = binPos + (size_t)bg * CAP;
    for (int j = 0; j < cnt; ++j)
      in_c += w_inT[(size_t)pl[j] * HH + hh];        // 16 lanes coalesce 64B

    v[idx]     = v_nw;
    i_st[idx]  = i_dec + rec + in_c;
    z_new[idx] = z;
  }
}

// K2: readout + LI cell.  out[t] uses io BEFORE adding this step's y
// (matches reference ordering).  One block, thread = (b,o).
__global__ void snn_step_k2(const float* __restrict__ z_new,  // [B,H]
                            const float* __restrict__ w_out,  // [OUT,H]
                            float* __restrict__ vo,           // [B,OUT]
                            float* __restrict__ io,           // [B,OUT]
                            float* __restrict__ out,          // [T,B,OUT]
                            int t) {
  const int tid = threadIdx.x;
  if (tid >= BB * OUTN) return;
  const int b = tid / OUTN, o = tid % OUTN;
  const float* zr = z_new + b * HH;
  const float* wr = w_out + o * HH;
  float y = 0.f;
  for (int h = 0; h < HH; ++h) y += zr[h] * wr[h];
  const float vold = vo[tid], iold = io[tid];
  const float vn = vold + 0.1f * (iold - vold);
  out[(size_t)t * BB * OUTN + tid] = vn;
  vo[tid] = vn;
  io[tid] = 0.8f * iold + y;
}

// ---------------- launch ----------------

extern "C" void kernel_launch(void* const* d_in, const int* in_sizes, int n_in,
                              void* d_out, int out_size, void* d_ws, size_t ws_size,
                              hipStream_t stream) {
  (void)in_sizes; (void)n_in; (void)out_size; (void)ws_size;
  const int*   x     = (const int*)d_in[0];    // [B,S,4] int32
  const float* w_in  = (const float*)d_in[1];  // [H,IN]
  const float* w_rec = (const float*)d_in[2];  // [H,H]
  const float* w_out = (const float*)d_in[3];  // [OUT,H]
  float* out = (float*)d_out;                  // [T,B,OUT]

  // workspace layout
  char* ws = (char*)d_ws;
  float* w_inT  = (float*)ws;                            // IN*H floats (8 MB)
  float* w_recT = w_inT  + (size_t)INN * HH;             // H*H floats (1 MB)
  float* v      = w_recT + (size_t)HH * HH;
  float* i_st   = v    + BB * HH;
  float* z0     = i_st + BB * HH;
  float* z1     = z0   + BB * HH;
  float* vo     = z1   + BB * HH;
  float* io     = vo   + BB * OUTN;
  int*   binCnt = (int*)(io + BB * OUTN);                // B*T ints
  int*   binPos = binCnt + BB * TSTEPS;                  // B*T*CAP ints (3 MB)

  // init state (v,i,z0,z1,vo,io are contiguous)
  const int nstate = 4 * BB * HH + 2 * BB * OUTN;
  snn_zero_kernel<<<(nstate + 255) / 256, 256, 0, stream>>>(v, nstate);

  snn_transpose_kernel<<<(HH * INN + 255) / 256, 256, 0, stream>>>(w_in, w_inT, HH, INN);
  snn_transpose_kernel<<<(HH * HH + 255) / 256, 256, 0, stream>>>(w_rec, w_recT, HH, HH);
  snn_bin_kernel<<<(BB * TSTEPS + 127) / 128, 128, 0, stream>>>(x, binPos, binCnt);

  for (int t = 0; t < TSTEPS; ++t) {
    const float* zp = (t & 1) ? z1 : z0;
    float*       zn = (t & 1) ? z0 : z1;
    snn_step_k1<<<8, 256, 0, stream>>>(w_recT, w_inT, binPos, binCnt,
                                       zp, zn, v, i_st, t);
    snn_step_k2<<<1, 320, 0, stream>>>(zn, w_out, vo, io, out, t);
  }
}